// TrueSVBlurOperator_53102975647806
// MI455X (gfx1250) — compile-verified
//
#include <hip/hip_runtime.h>
#include <stdint.h>

#define H_IMG 512
#define W_IMG 512
#define C_IMG 3
#define KH 9
#define KW 9
#define TAPS 81
#define PAD 4
#define TILE_W 128
#define XCOLS (TILE_W + 2 * PAD)          // 136
#define K_FLOATS (TILE_W * TAPS)          // 10368 floats = 41472 B (contiguous)
#define X_FLOATS (C_IMG * KH * XCOLS)     // 3672 floats = 14688 B

// ---------- CDNA5 feature detection (device pass only) ----------
#if defined(__AMDGCN__)
#if __has_builtin(__builtin_amdgcn_global_load_async_to_lds_b32) &&   \
    __has_builtin(__builtin_amdgcn_s_wait_asynccnt)
#define HAS_ASYNC 1
#endif
#if __has_builtin(__builtin_amdgcn_tensor_load_to_lds) &&             \
    __has_builtin(__builtin_amdgcn_s_wait_tensorcnt)
#define HAS_TDM 1
#endif
#endif

typedef uint32_t u32x4 __attribute__((ext_vector_type(4)));
typedef int      i32x8 __attribute__((ext_vector_type(8)));
typedef int      i32x4 __attribute__((ext_vector_type(4)));

// jnp.pad mode="reflect" (no edge repeat): -1 -> 1, n -> n-2
__device__ __forceinline__ int reflect_idx(int i, int n) {
  i = (i < 0) ? -i : i;
  if (i >= n) i = 2 * n - 2 - i;
  return i;
}

__global__ __launch_bounds__(TILE_W) void svblur_kernel(
    const float* __restrict__ x,
    const float* __restrict__ kf,
    float* __restrict__ out) {
  __shared__ __align__(16) float sK[K_FLOATS];
  __shared__ __align__(16) float sX[X_FLOATS];

  const int tid = threadIdx.x;
  const int blk = blockIdx.x;
  const int h  = blk >> 2;                 // 4 tiles per image row
  const int w0 = (blk & 3) * TILE_W;

  const float* Kblk = kf + (size_t)(h * W_IMG + w0) * TAPS;

  // ---------------- Stage K (41472 contiguous bytes) ----------------
#if defined(HAS_TDM)
  if (tid < 32) {  // one wave issues the TDM descriptor (EXEC ignored by TDM)
    const uint32_t lds_addr = (uint32_t)(uintptr_t)(&sK[0]);
    const uint64_t gaddr = (uint64_t)(uintptr_t)Kblk;
    // D# group0: count=1 | lds_addr | global_addr[56:0] | type=2
    u32x4 g0 = {
        0x1u,
        lds_addr,
        (uint32_t)(gaddr & 0xFFFFFFFFu),
        (uint32_t)((gaddr >> 32) & 0x1FFFFFFu) | (2u << 30)};
    // D# group1: data_size=4B; 1-D tile: tensor_dim0=tile_dim0=K_FLOATS,
    // tensor_dim1=1, tile_dim1/2 unused, dim0 stride = K_FLOATS
    i32x8 g1 = {
        (int)(2u << 16),                                   // data_size = 4B
        (int)((K_FLOATS & 0xFFFFu) << 16),                 // tensor_dim0 lo16
        (int)(((K_FLOATS >> 16) & 0xFFFFu) | (1u << 16)),  // dim0 hi16 | tensor_dim1 lo16 = 1
        (int)((K_FLOATS & 0xFFFFu) << 16),                 // tensor_dim1 hi=0 | tile_dim0
        0,                                                 // tile_dim1=0, tile_dim2=0
        (int)K_FLOATS,                                     // tensor_dim0_stride lo32
        0, 0};
    i32x4 gz = {0, 0, 0, 0};
#if __clang_major__ >= 23
    i32x8 gz8 = {0, 0, 0, 0, 0, 0, 0, 0};
    __builtin_amdgcn_tensor_load_to_lds(g0, g1, gz, gz, gz8, 0);
#else
    __builtin_amdgcn_tensor_load_to_lds(g0, g1, gz, gz, 0);
#endif
  }
#elif defined(HAS_ASYNC)
  for (int t = 0; t < TAPS; ++t) {  // coalesced b32 async copies
    const int idx = t * TILE_W + tid;
    __builtin_amdgcn_global_load_async_to_lds_b32(
        (int*)(uintptr_t)(Kblk + idx), (int*)&sK[idx], 0, 0);
  }
#else
  for (int idx = tid; idx < K_FLOATS / 4; idx += TILE_W) {
    float4 v = *reinterpret_cast<const float4*>(Kblk + 4 * idx);
    *reinterpret_cast<float4*>(&sK[4 * idx]) = v;
  }
#endif

  // ------------- Stage x halo tile (reflect-padded gather) -------------
#if defined(HAS_ASYNC)
  for (int e = tid; e < X_FLOATS; e += TILE_W) {
    const int c   = e / (KH * XCOLS);
    const int rem = e - c * (KH * XCOLS);
    const int r   = rem / XCOLS;
    const int col = rem - r * XCOLS;
    const int sr  = reflect_idx(h + r - PAD, H_IMG);
    const int sc  = reflect_idx(w0 + col - PAD, W_IMG);
    const float* gp = x + ((size_t)c * H_IMG + sr) * W_IMG + sc;
    __builtin_amdgcn_global_load_async_to_lds_b32(
        (int*)(uintptr_t)gp, (int*)&sX[e], 0, 0);
  }
  __builtin_amdgcn_s_wait_asynccnt(0);
#else
  for (int e = tid; e < X_FLOATS; e += TILE_W) {
    const int c   = e / (KH * XCOLS);
    const int rem = e - c * (KH * XCOLS);
    const int r   = rem / XCOLS;
    const int col = rem - r * XCOLS;
    const int sr  = reflect_idx(h + r - PAD, H_IMG);
    const int sc  = reflect_idx(w0 + col - PAD, W_IMG);
    sX[e] = x[((size_t)c * H_IMG + sr) * W_IMG + sc];
  }
#endif
#if defined(HAS_TDM)
  if (tid < 32) __builtin_amdgcn_s_wait_tensorcnt(0);
#endif
  __syncthreads();

  // ------------- Compute: 81 taps x 3 channels from LDS -------------
  float a0 = 0.f, a1 = 0.f, a2 = 0.f;
  const float* kp = &sK[tid * TAPS];  // lane stride 81 words = 17 mod 64 banks: conflict-free
#pragma unroll
  for (int i = 0; i < KH; ++i) {
#pragma unroll
    for (int j = 0; j < KW; ++j) {
      const float k = kp[i * KW + j];
      a0 += sX[(0 * KH + i) * XCOLS + tid + j] * k;
      a1 += sX[(1 * KH + i) * XCOLS + tid + j] * k;
      a2 += sX[(2 * KH + i) * XCOLS + tid + j] * k;
    }
  }

  const size_t o = (size_t)h * W_IMG + w0 + tid;
  out[o]                             = a0;
  out[o + (size_t)H_IMG * W_IMG]     = a1;
  out[o + 2 * (size_t)H_IMG * W_IMG] = a2;
}

extern "C" void kernel_launch(void* const* d_in, const int* in_sizes, int n_in,
                              void* d_out, int out_size, void* d_ws, size_t ws_size,
                              hipStream_t stream) {
  (void)in_sizes; (void)n_in; (void)out_size; (void)d_ws; (void)ws_size;
  const float* x  = (const float*)d_in[0];   // (1,3,512,512) fp32
  const float* kf = (const float*)d_in[1];   // (512,512,9,9) fp32
  float* out = (float*)d_out;                // (1,3,512,512) fp32
  dim3 grid(H_IMG * (W_IMG / TILE_W));       // 512 rows * 4 tiles = 2048 blocks
  dim3 block(TILE_W);                        // 128 threads = 4 waves (wave32)
  svblur_kernel<<<grid, block, 0, stream>>>(x, kf, out);
}